// FM_38560216383900
// MI455X (gfx1250) — compile-verified
//
#include <hip/hip_runtime.h>

typedef __attribute__((ext_vector_type(2))) float v2f;
typedef __attribute__((ext_vector_type(8))) float v8f;

#define NF      8
#define VOCABSZ 12
#define DIM     64
#define BATCH   131072
#define NROWS   (NF * VOCABSZ)   // 96 table rows
#define LSTRIDE 66               // padded LDS row stride (floats) to break 256B bank stride
#define NTILES  (BATCH / 16)     // 8192 wave tiles

__global__ void fm_zero_loss(float* p) {
    if (threadIdx.x == 0 && blockIdx.x == 0) *p = 0.0f;
}

__global__ __launch_bounds__(256)
void fm_fused_kernel(const float* __restrict__ embT,   // (8,12,64)
                     const float* __restrict__ embF,   // (8,12,1)
                     const float* __restrict__ actE,   // (2,64)
                     const float* __restrict__ actF,   // (2,1)
                     const float* __restrict__ label,  // (B,2)
                     const float* __restrict__ posw,   // (B,2)
                     const int*   __restrict__ feats,  // (8,B)
                     float* __restrict__ out_inf,      // (B,2)
                     float*              out_loss,     // scalar
                     float* __restrict__ out_pair)     // (B,1)
{
    __shared__ __align__(16) float sT[NROWS * LSTRIDE]; // ~25.3 KB padded tables
    __shared__ float sN2[NROWS];                        // per-(f,v) squared norms
    __shared__ float sF[NROWS];                         // first-order table
    __shared__ float sA[2 * DIM];                       // action embeddings
    __shared__ float sAF[2];                            // action first-order

    const int tid = threadIdx.x;

    for (int i = tid; i < NROWS * DIM; i += 256) {
        int r = i >> 6, c = i & 63;
        sT[r * LSTRIDE + c] = embT[i];
    }
    for (int i = tid; i < NROWS; i += 256) sF[i] = embF[i];
    for (int i = tid; i < 2 * DIM; i += 256) sA[i] = actE[i];
    if (tid < 2) sAF[tid] = actF[tid];
    __syncthreads();

    for (int i = tid; i < NROWS; i += 256) {
        const float* row = &sT[i * LSTRIDE];
        float acc = 0.f;
        #pragma unroll
        for (int k = 0; k < DIM; ++k) acc += row[k] * row[k];
        sN2[i] = acc;
    }
    __syncthreads();

    const int lane  = tid & 31;
    const int hi    = lane >> 4;   // K-half (A frag) / row-half (C frag)
    const int n     = lane & 15;   // A-frag row owned by this lane; C-frag column
    const int gwave = blockIdx.x * 8 + (tid >> 5);
    const int nwv   = gridDim.x * 8;

    // B fragment: B[k][col] = actE[col*64+k] for col<2 else 0. Constant per lane.
    v2f bfrag[16];
    #pragma unroll
    for (int kc = 0; kc < 16; ++kc) {
        const int k0 = 4 * kc + 2 * hi;
        bfrag[kc].x = (n < 2) ? sA[n * DIM + k0]     : 0.f;
        bfrag[kc].y = (n < 2) ? sA[n * DIM + k0 + 1] : 0.f;
    }
    const float af = (n < 2) ? sAF[n] : 0.f;

    float loss_acc = 0.f;

    for (int tile = gwave; tile < NTILES; tile += nwv) {
        const int row_g = tile * 16 + n;   // row this lane gathers (dup across halves)

        int   vbase[NF];
        float fo = 0.f, n2 = 0.f;
        #pragma unroll
        for (int f = 0; f < NF; ++f) {
            const int v   = feats[f * BATCH + row_g];
            const int idx = f * VOCABSZ + v;
            vbase[f] = idx * LSTRIDE;
            fo += sF[idx];
            n2 += sN2[idx];
        }

        // Sweep K: build S components in A-fragment order, accumulate WMMA + ||S||^2
        v8f c = {0.f, 0.f, 0.f, 0.f, 0.f, 0.f, 0.f, 0.f};
        float ssp = 0.f;
        #pragma unroll
        for (int kc = 0; kc < 16; ++kc) {
            const int k0 = 4 * kc + 2 * hi;
            float sx = 0.f, sy = 0.f;
            #pragma unroll
            for (int f = 0; f < NF; ++f) {
                const float2 e = *(const float2*)&sT[vbase[f] + k0];
                sx += e.x; sy += e.y;
            }
            ssp += sx * sx + sy * sy;
            v2f a; a.x = sx; a.y = sy;
            c = __builtin_amdgcn_wmma_f32_16x16x4_f32(
                    false, a, false, bfrag[kc], (short)0, c, false, false);
        }

        // pair: combine the two K-halves of ||S||^2
        const float ss = ssp + __shfl_xor(ssp, 16, 32);
        if (hi == 0) out_pair[row_g] = 0.5f * (ss - n2);

        // inferences + loss: lane (hi, n<2) holds D[8*hi + r][n] in c[r]
        #pragma unroll
        for (int r = 0; r < 8; ++r) {
            const float fo_r = __shfl(fo, r + 8 * hi, 32);
            if (n < 2) {
                const int   row = tile * 16 + 8 * hi + r;
                const float inf = c[r] + fo_r + af;
                out_inf[row * 2 + n] = inf;
                const float d = inf - label[row * 2 + n];
                loss_acc += posw[row * 2 + n] * d * d;
            }
        }
    }

    if (n < 2) atomicAdd(out_loss, loss_acc * (1.0f / (float)BATCH));
}

extern "C" void kernel_launch(void* const* d_in, const int* in_sizes, int n_in,
                              void* d_out, int out_size, void* d_ws, size_t ws_size,
                              hipStream_t stream) {
    const float* embT  = (const float*)d_in[0];
    const float* embF  = (const float*)d_in[1];
    const float* actE  = (const float*)d_in[2];
    const float* actF  = (const float*)d_in[3];
    const float* label = (const float*)d_in[4];
    const float* posw  = (const float*)d_in[5];
    const int*   feats = (const int*)  d_in[6];

    float* out      = (float*)d_out;
    float* out_inf  = out;                 // BATCH*2
    float* out_loss = out + BATCH * 2;     // 1
    float* out_pair = out + BATCH * 2 + 1; // BATCH

    fm_zero_loss<<<1, 32, 0, stream>>>(out_loss);
    fm_fused_kernel<<<512, 256, 0, stream>>>(embT, embF, actE, actF, label, posw,
                                             feats, out_inf, out_loss, out_pair);
}